// DynamicFilterConvLayer_8796093022262
// MI455X (gfx1250) — compile-verified
//
#include <hip/hip_runtime.h>

// DynamicFilterConv for MI455X (gfx1250, wave32).
// Per location: out[f] = sum_p patch[p] * filt[f,p]  (16x144 matvec, dynamic filter)
// Memory-bound: filters stream = 231 MB, floor ~10us @ 23.3 TB/s.
// One wave = one location; V_WMMA_F32_16X16X4_F32 with broadcast-B columns.

typedef float v2f __attribute__((ext_vector_type(2)));
typedef float v8f __attribute__((ext_vector_type(8)));

#define KS 3
#define NF 16
#define CH 16
#define HH 56
#define WW 56
#define KKC 144               // 3*3*16 patch length
#define WAVES_PER_BLOCK 8

__global__ __launch_bounds__(256) void dynconv_wmma_kernel(
    const float* __restrict__ inpt,     // [B,56,56,16]
    const float* __restrict__ filters,  // [B,56,56,2304]
    float* __restrict__ out,            // [B,56,56,16]
    int nloc)
{
  __shared__ float patch_lds[WAVES_PER_BLOCK * KKC];

  const int tid  = (int)threadIdx.x;
  const int lane = tid & 31;
  const int wave = tid >> 5;
  const int loc  = (int)blockIdx.x * WAVES_PER_BLOCK + wave;   // wave-uniform
  if (loc >= nloc) return;                                     // whole wave exits together

  const int b  = loc / (HH * WW);
  const int hw = loc - b * HH * WW;
  const int h  = hw / WW;
  const int w  = hw - h * WW;

  // ---- stage the 144-float im2col patch (order ki,kj,c with c fastest) in LDS.
  // Per-wave region: only this wave reads it -> same-wave DS ordering, no barrier.
  float* pl = &patch_lds[wave * KKC];
  #pragma unroll
  for (int r = 0; r < 5; ++r) {
    int p = lane + 32 * r;
    if (p < KKC) {
      int cc = p & 15;
      int pk = p >> 4;                // ki*3 + kj, 0..8
      int ki = pk / 3;
      int kj = pk - 3 * ki;
      int hi = h + ki - 1;
      int wi = w + kj - 1;
      float v = 0.0f;
      if (hi >= 0 && hi < HH && wi >= 0 && wi < WW)
        v = inpt[((size_t)(b * HH + hi) * WW + wi) * CH + cc];
      pl[p] = v;
    }
  }

  // ---- per-lane base pointers matching WMMA f32 16x16x4 register layouts.
  // A (16x4, MxK): lanes 0-15 hold M=lane, {K0,K1}; lanes 16-31 hold M=lane-16, {K2,K3}.
  //   -> one b64 load per chunk: filt[loc, lane%16, 4k + 2*half + {0,1}]
  // B (4x16, KxN) broadcast: VGPR0 = K0 (lanes 0-15) / K2 (lanes 16-31), VGPR1 = K1/K3,
  //   same value in every column -> one ds_load_b64 of patch[4k + 2*half + {0,1}].
  const int half = lane >> 4;
  const float* fbase = filters + (size_t)loc * (NF * KKC)
                               + (size_t)(lane & 15) * KKC + (half << 1);
  const float* pbase = pl + (half << 1);

  v8f acc0 = {0.f,0.f,0.f,0.f,0.f,0.f,0.f,0.f};
  v8f acc1 = {0.f,0.f,0.f,0.f,0.f,0.f,0.f,0.f};

  #pragma unroll
  for (int k = 0; k < KKC / 4; ++k) {        // 36 chunks of K=4
    v2f a  = __builtin_nontemporal_load((const v2f*)(fbase + 4 * k)); // single-use stream: NT
    v2f bv = *(const v2f*)(pbase + 4 * k);
    if (k & 1)
      acc1 = __builtin_amdgcn_wmma_f32_16x16x4_f32(false, a, false, bv,
                                                   (short)0, acc1, false, false);
    else
      acc0 = __builtin_amdgcn_wmma_f32_16x16x4_f32(false, a, false, bv,
                                                   (short)0, acc0, false, false);
  }
  v8f acc = acc0 + acc1;

  // ---- D (16x16): lane n (0-15), VGPR r -> (M=r, N=n); lanes 16-31 -> M=r+8.
  // All columns identical: lane 0 holds out[0..7], lane 16 holds out[8..15].
  if ((lane & 15) == 0) {
    float* op = out + (size_t)loc * NF + half * 8;
    *(float4*)(op + 0) = make_float4(acc[0], acc[1], acc[2], acc[3]);
    *(float4*)(op + 4) = make_float4(acc[4], acc[5], acc[6], acc[7]);
  }
}

extern "C" void kernel_launch(void* const* d_in, const int* in_sizes, int n_in,
                              void* d_out, int out_size, void* d_ws, size_t ws_size,
                              hipStream_t stream) {
  const float* inpt    = (const float*)d_in[0];  // [8,56,56,16]  f32
  const float* filters = (const float*)d_in[1];  // [8,56,56,2304] f32
  float* out           = (float*)d_out;          // [8,56,56,16]  f32

  const int nloc = in_sizes[0] / CH;             // B*H*W = 25088
  dim3 block(256);                               // 8 waves -> 8 locations per block
  dim3 grid((nloc + WAVES_PER_BLOCK - 1) / WAVES_PER_BLOCK);
  hipLaunchKernelGGL(dynconv_wmma_kernel, grid, block, 0, stream,
                     inpt, filters, out, nloc);
}